// DeformableResidualBlock_68109591381035
// MI455X (gfx1250) — compile-verified
//
#include <hip/hip_runtime.h>
#include <hip/hip_bf16.h>

// ---------------------------------------------------------------------------
// DeformableResidualBlock for MI455X (gfx1250, wave32).
//   - offset conv:     WMMA GEMM [18(pad 32) x 576] x [576 x 65536]
//   - deformable conv: WMMA GEMM [64 x 576] x [576 x 65536], bilinear B,
//                      A-slices staged to LDS via TENSOR_LOAD_TO_LDS (TDM),
//                      double-buffered, consumed with ds_load_b128 (explicit
//                      addrspace(3) so the loads stay on the DS pipe).
// ---------------------------------------------------------------------------

typedef __attribute__((ext_vector_type(2))) float        v2f;
typedef __attribute__((ext_vector_type(4))) float        v4f;
typedef __attribute__((ext_vector_type(8))) float        v8f;
typedef __attribute__((ext_vector_type(4))) unsigned int u32x4;
typedef __attribute__((ext_vector_type(4))) int          i32x4;
typedef __attribute__((ext_vector_type(8))) int          i32x8;

// LDS-resident vector pointer: forces ds_load_b128 (no flat fallback).
typedef __attribute__((address_space(3))) const v4f* lds_cv4f;

#define BB 4
#define CC 64
#define HH 128
#define WW 128
#define HWSZ (HH * WW)          // 16384
#define CHW (CC * HWSZ)         // 1048576
#define KTAPS 9
#define KDIM (CC * KTAPS)       // 576
#define NPIX (BB * HWSZ)        // 65536
#define OFFC (2 * KTAPS)        // 18 offset channels

// ---------------------------------------------------------------------------
// TDM: issue a 2-D tensor tile load (4-byte elements) into LDS.
// D# packing per CDNA5 ISA:
//   group0: count=1 | lds_addr | global_addr(57b) | type=2
//   group1: wg_mask=0, data_size=2(4B), dims/strides as below, no padding.
// Tracked by TENSORcnt; pair with __builtin_amdgcn_s_wait_tensorcnt.
// ---------------------------------------------------------------------------
__device__ __forceinline__ void tdm_load_2d_f32(unsigned lds_byte_off,
                                                const float* gsrc,
                                                unsigned tile_d0, unsigned tile_d1,
                                                unsigned tensor_d0, unsigned tensor_d1,
                                                unsigned stride0) {
    unsigned long long ga = (unsigned long long)gsrc;
    u32x4 g0;
    g0.x = 1u;                                        // count=1, user descriptor
    g0.y = lds_byte_off;                              // LDS destination (bytes)
    g0.z = (unsigned)(ga & 0xffffffffu);              // global_addr[31:0]
    g0.w = (unsigned)((ga >> 32) & 0x01ffffffu)       // global_addr[56:32]
         | (2u << 30);                                // type=2 ("image")
    i32x8 g1;
    g1[0] = (int)(2u << 16);                          // wg_mask=0, data_size=4B
    g1[1] = (int)((tensor_d0 & 0xffffu) << 16);       // tensor_dim0[15:0] @bit48
    g1[2] = (int)((tensor_d0 >> 16) | ((tensor_d1 & 0xffffu) << 16));
    g1[3] = (int)((tensor_d1 >> 16) | (tile_d0 << 16));   // tile_dim0 @bit112
    g1[4] = (int)(tile_d1 & 0xffffu);                 // tile_dim1, tile_dim2=0
    g1[5] = (int)stride0;                             // tensor_dim0_stride[31:0]
    g1[6] = 0;                                        // stride0 hi, stride1 lo
    g1[7] = 0;                                        // stride1 hi
    i32x4 z4 = {0, 0, 0, 0};
#if defined(__clang_major__) && (__clang_major__ >= 23)
    i32x8 z8 = {0, 0, 0, 0, 0, 0, 0, 0};
    __builtin_amdgcn_tensor_load_to_lds(g0, g1, z4, z4, z8, 0);
#else
    __builtin_amdgcn_tensor_load_to_lds(g0, g1, z4, z4, 0);
#endif
}

// ---------------------------------------------------------------------------
// Deform weights: w[m][c][ky][kx] (64x64x3x3) -> wTi[kk][(m&15)*4 + (m>>4)],
// kk = k*64 + c. Lane reading col=m&15 gets 4 Cout-tile elements of a K-row
// as one aligned float4 (ds_load_b128 after TDM staging).
// ---------------------------------------------------------------------------
__global__ void transpose_w_kernel(const float* __restrict__ w,
                                   float* __restrict__ wTi) {
    int idx = blockIdx.x * 256 + threadIdx.x;          // 0 .. 36863
    if (idx >= KDIM * CC) return;
    int m  = idx & 63;
    int kk = idx >> 6;
    int c  = kk & 63;
    int k  = kk >> 6;
    wTi[kk * 64 + (m & 15) * 4 + (m >> 4)] = w[m * KDIM + c * KTAPS + k];
}

// Offset-conv weights: ow (18x64x3x3) -> owTi[kk][(m&15)*2+(m>>4)], m pad->32.
__global__ void transpose_ow_kernel(const float* __restrict__ ow,
                                    float* __restrict__ owTi) {
    int idx = blockIdx.x * 256 + threadIdx.x;          // 0 .. 18431
    if (idx >= KDIM * 32) return;
    int m  = idx & 31;
    int kk = idx >> 5;
    int c  = kk & 63;
    int k  = kk >> 6;
    float v = (m < OFFC) ? ow[m * KDIM + c * KTAPS + k] : 0.f;
    owTi[kk * 32 + (m & 15) * 2 + (m >> 4)] = v;
}

// ---------------------------------------------------------------------------
// Offset conv on the matrix pipe. One wave32 per 16-pixel tile, 2 Cout tiles.
// ---------------------------------------------------------------------------
__global__ void offset_conv_wmma_kernel(const float* __restrict__ x,
                                        const float* __restrict__ owTi,
                                        const float* __restrict__ ob,
                                        float* __restrict__ off) {
    const int lane = threadIdx.x & 31;
    const int wave = threadIdx.x >> 5;
    const int tile = blockIdx.x * 8 + wave;            // 0 .. 4095
    const int col  = lane & 15;
    const bool hi  = (lane >= 16);

    const int pid = tile * 16 + col;
    const int b   = pid >> 14;
    const int sp  = pid & (HWSZ - 1);
    const int py  = sp >> 7;
    const int px  = sp & (WW - 1);

    const float* xb = x + b * CHW;
    const v2f*   aw = (const v2f*)owTi;                // 2 floats per (kk, col)

    v8f acc0 = {}, acc1 = {};

    for (int k = 0; k < KTAPS; ++k) {
        const int yy = py + k / 3 - 1;
        const int xx = px + k % 3 - 1;
        const float vf =
            ((unsigned)yy < (unsigned)HH && (unsigned)xx < (unsigned)WW) ? 1.f : 0.f;
        const int cyy = yy < 0 ? 0 : (yy > HH - 1 ? HH - 1 : yy);
        const int cxx = xx < 0 ? 0 : (xx > WW - 1 ? WW - 1 : xx);
        const int idx = cyy * WW + cxx;

        const int kkbase = k * CC;
        for (int s = 0; s < 16; ++s) {
            const int c0 = s * 4 + (hi ? 2 : 0);
            v2f bfrag;
            bfrag.x = vf * xb[(c0 + 0) * HWSZ + idx];
            bfrag.y = vf * xb[(c0 + 1) * HWSZ + idx];

            const int kk0 = kkbase + c0;
            v2f r0 = aw[kk0 * 16 + col];               // row kk0: tiles {0,1}
            v2f r1 = aw[(kk0 + 1) * 16 + col];         // row kk0+1
            v2f a0 = { r0.x, r1.x };
            v2f a1 = { r0.y, r1.y };

            acc0 = __builtin_amdgcn_wmma_f32_16x16x4_f32(false, a0, false, bfrag,
                                                         (short)0, acc0, false, false);
            acc1 = __builtin_amdgcn_wmma_f32_16x16x4_f32(false, a1, false, bfrag,
                                                         (short)0, acc1, false, false);
        }
    }

    float* ob_out = off + b * (OFFC * HWSZ);
#pragma unroll
    for (int v = 0; v < 8; ++v) {
        const int mr = v + (hi ? 8 : 0);
        ob_out[mr * HWSZ + sp] = acc0[v] + ob[mr];     // rows 0..15
        if (mr < 2)
            ob_out[(16 + mr) * HWSZ + sp] = acc1[v] + ob[16 + mr];  // rows 16,17
    }
}

// ---------------------------------------------------------------------------
// Deformable sampling + WMMA GEMM. One wave32 per 16-pixel tile, 4 Cout tiles.
// Weight slices (16 KB per tap) are DMA'd into LDS by the TDM, double-buffered,
// so the VMEM pipe is dedicated to the bilinear gathers.
// ---------------------------------------------------------------------------
__global__ void deform_wmma_kernel(const float* __restrict__ x,
                                   const float* __restrict__ off,
                                   const float* __restrict__ wTi,
                                   const float* __restrict__ bias,
                                   float* __restrict__ y) {
    __shared__ float lw[2][CC * CC];                   // 2 x 16 KB tap slices

    const int lane = threadIdx.x & 31;
    const int wave = threadIdx.x >> 5;
    const int tile = blockIdx.x * 8 + wave;            // 0 .. 4095
    const int col  = lane & 15;
    const bool hi  = (lane >= 16);

    const int pid = tile * 16 + col;
    const int b   = pid >> 14;
    const int sp  = pid & (HWSZ - 1);
    const int py  = sp >> 7;
    const int px  = sp & (WW - 1);

    const float* xb   = x + b * CHW;
    const float* offb = off + b * (OFFC * HWSZ);

    v8f acc0 = {}, acc1 = {}, acc2 = {}, acc3 = {};

    // Prologue: wave 0 kicks off the tap-0 weight slice DMA. `lw` is the only
    // LDS allocation in this kernel, so its LDS byte offset is 0.
    if (threadIdx.x < 32) {
        tdm_load_2d_f32(0u, wTi, CC, CC, CC, KDIM, CC);
    }

    for (int k = 0; k < KTAPS; ++k) {
        // Issue next tap's DMA, then wait for the current slice to land.
        if (threadIdx.x < 32) {
            if (k + 1 < KTAPS) {
                tdm_load_2d_f32((unsigned)(((k + 1) & 1) * (CC * CC * 4)),
                                wTi + (k + 1) * (CC * CC), CC, CC, CC, KDIM, CC);
                __builtin_amdgcn_s_wait_tensorcnt(1);
            } else {
                __builtin_amdgcn_s_wait_tensorcnt(0);
            }
        }
        __syncthreads();                               // slice k visible to all

        // --- per-tap bilinear setup (hoisted over 16 WMMA steps) ---
        const int tdy = k / 3 - 1;
        const int tdx = k % 3 - 1;
        float cy = (float)(py + tdy) + offb[(2 * k + 0) * HWSZ + sp];
        float cx = (float)(px + tdx) + offb[(2 * k + 1) * HWSZ + sp];
        if (k + 1 < KTAPS)
            __builtin_prefetch(&offb[(2 * k + 2) * HWSZ + sp], 0, 0);
        float fy = floorf(cy), fx = floorf(cx);
        float wy = cy - fy,    wx = cx - fx;
        int y0 = (int)fy, x0 = (int)fx;
        int y1 = y0 + 1,  x1 = x0 + 1;

        float vy0 = ((unsigned)y0 < (unsigned)HH) ? 1.f : 0.f;
        float vy1 = ((unsigned)y1 < (unsigned)HH) ? 1.f : 0.f;
        float vx0 = ((unsigned)x0 < (unsigned)WW) ? 1.f : 0.f;
        float vx1 = ((unsigned)x1 < (unsigned)WW) ? 1.f : 0.f;

        float w00 = (1.f - wy) * (1.f - wx) * vy0 * vx0;
        float w01 = (1.f - wy) * wx         * vy0 * vx1;
        float w10 = wy * (1.f - wx)         * vy1 * vx0;
        float w11 = wy * wx                 * vy1 * vx1;

        int cy0 = y0 < 0 ? 0 : (y0 > HH - 1 ? HH - 1 : y0);
        int cy1 = y1 < 0 ? 0 : (y1 > HH - 1 ? HH - 1 : y1);
        int cx0 = x0 < 0 ? 0 : (x0 > WW - 1 ? WW - 1 : x0);
        int cx1 = x1 < 0 ? 0 : (x1 > WW - 1 ? WW - 1 : x1);
        int i00 = cy0 * WW + cx0;
        int i01 = cy0 * WW + cx1;
        int i10 = cy1 * WW + cx0;
        int i11 = cy1 * WW + cx1;

        // Explicit addrspace(3) pointer: guarantees ds_load_b128 on the DS pipe
        // (a generic-pointer select over buffers was lowering to flat loads).
        lds_cv4f lwv = (lds_cv4f)(&lw[k & 1][0]);

        // --- 16 WMMA steps sweep the 64 channels of this tap ---
        for (int s = 0; s < 16; ++s) {
            // B fragment (4x16 f32): lanes 0-15 carry K rows {cb,cb+1},
            // lanes 16-31 carry {cb+2,cb+3}, at column N = lane&15.
            const int c0 = s * 4 + (hi ? 2 : 0);
            const float* p0 = xb + c0 * HWSZ;
            const float* p1 = p0 + HWSZ;
            v2f bfrag;
            bfrag.x = w00 * p0[i00] + w01 * p0[i01] + w10 * p0[i10] + w11 * p0[i11];
            bfrag.y = w00 * p1[i00] + w01 * p1[i01] + w10 * p1[i10] + w11 * p1[i11];

            // A fragments: one ds_load_b128 per K-row gives all 4 Cout tiles.
            v4f r0 = lwv[(c0 + 0) * 16 + col];
            v4f r1 = lwv[(c0 + 1) * 16 + col];
            v2f a0 = { r0.x, r1.x };
            v2f a1 = { r0.y, r1.y };
            v2f a2 = { r0.z, r1.z };
            v2f a3 = { r0.w, r1.w };

            acc0 = __builtin_amdgcn_wmma_f32_16x16x4_f32(false, a0, false, bfrag,
                                                         (short)0, acc0, false, false);
            acc1 = __builtin_amdgcn_wmma_f32_16x16x4_f32(false, a1, false, bfrag,
                                                         (short)0, acc1, false, false);
            acc2 = __builtin_amdgcn_wmma_f32_16x16x4_f32(false, a2, false, bfrag,
                                                         (short)0, acc2, false, false);
            acc3 = __builtin_amdgcn_wmma_f32_16x16x4_f32(false, a3, false, bfrag,
                                                         (short)0, acc3, false, false);
        }
        __syncthreads();   // all waves done with slice k before it is reused
    }

    // D layout: VGPR v, lane -> row M = v + (hi?8:0), col N = lane&15.
    float* yb = y + b * CHW;
#pragma unroll
    for (int v = 0; v < 8; ++v) {
        const int mr = v + (hi ? 8 : 0);
        yb[(0 * 16 + mr) * HWSZ + sp] = acc0[v] + bias[0 * 16 + mr];
        yb[(1 * 16 + mr) * HWSZ + sp] = acc1[v] + bias[1 * 16 + mr];
        yb[(2 * 16 + mr) * HWSZ + sp] = acc2[v] + bias[2 * 16 + mr];
        yb[(3 * 16 + mr) * HWSZ + sp] = acc3[v] + bias[3 * 16 + mr];
    }
}

// ---------------------------------------------------------------------------
// GroupNorm(1) stats, pass 1: deterministic block partial sums (LDS reduce).
// ---------------------------------------------------------------------------
__global__ void reduce_kernel(const float* __restrict__ y,
                              float* __restrict__ partials) {
    const int b     = blockIdx.x >> 7;
    const int chunk = blockIdx.x & 127;
    const float* p = y + b * CHW + chunk * 8192;

    float s = 0.f, ss = 0.f;
    for (int i = threadIdx.x; i < 8192; i += 256) {
        float v = p[i];
        s += v;
        ss = fmaf(v, v, ss);
    }
    __shared__ float sh0[256];
    __shared__ float sh1[256];
    sh0[threadIdx.x] = s;
    sh1[threadIdx.x] = ss;
    __syncthreads();
    for (int st = 128; st > 0; st >>= 1) {
        if (threadIdx.x < st) {
            sh0[threadIdx.x] += sh0[threadIdx.x + st];
            sh1[threadIdx.x] += sh1[threadIdx.x + st];
        }
        __syncthreads();
    }
    if (threadIdx.x == 0) {
        partials[blockIdx.x * 2 + 0] = sh0[0];
        partials[blockIdx.x * 2 + 1] = sh1[0];
    }
}

// Pass 2: fixed-order serial sum -> mean, rsqrt(var+eps).
__global__ void stats_kernel(const float* __restrict__ partials,
                             float* __restrict__ stats) {
    int b = threadIdx.x;
    if (b >= BB) return;
    float s = 0.f, ss = 0.f;
    for (int i = 0; i < 128; ++i) {
        s  += partials[(b * 128 + i) * 2 + 0];
        ss += partials[(b * 128 + i) * 2 + 1];
    }
    const float invN = 1.f / (float)CHW;
    float m = s * invN;
    float var = ss * invN - m * m;
    stats[b]     = m;
    stats[4 + b] = rsqrtf(var + 1e-5f);
}

// ---------------------------------------------------------------------------
// Fused GN(affine) + ReLU + residual add.
// ---------------------------------------------------------------------------
__global__ void gn_relu_res_kernel(const float* __restrict__ y,
                                   const float* __restrict__ xin,
                                   const float* __restrict__ gamma,
                                   const float* __restrict__ beta,
                                   const float* __restrict__ stats,
                                   float* __restrict__ out) {
    int idx = blockIdx.x * 256 + threadIdx.x;          // 0 .. B*C*H*W-1
    int b = idx >> 20;                                 // /CHW
    int c = (idx >> 14) & (CC - 1);
    float m   = stats[b];
    float inv = stats[4 + b];
    float v = (y[idx] - m) * inv * gamma[c] + beta[c];
    v = v > 0.f ? v : 0.f;
    out[idx] = v + xin[idx];
}

// ---------------------------------------------------------------------------
// Host-side orchestration
// ---------------------------------------------------------------------------
static void run_block(const float* xin, const float* w, const float* bias,
                      const float* ow, const float* ob,
                      const float* gamma, const float* beta,
                      float* off, float* ybuf, float* wTi, float* owTi,
                      float* partials, float* stats,
                      float* out, hipStream_t stream) {
    transpose_w_kernel<<<(KDIM * CC + 255) / 256, 256, 0, stream>>>(w, wTi);
    transpose_ow_kernel<<<(KDIM * 32 + 255) / 256, 256, 0, stream>>>(ow, owTi);
    offset_conv_wmma_kernel<<<NPIX / 16 / 8, 256, 0, stream>>>(xin, owTi, ob, off);
    deform_wmma_kernel<<<NPIX / 16 / 8, 256, 0, stream>>>(xin, off, wTi, bias, ybuf);
    reduce_kernel<<<BB * 128, 256, 0, stream>>>(ybuf, partials);
    stats_kernel<<<1, 32, 0, stream>>>(partials, stats);
    gn_relu_res_kernel<<<(BB * CHW) / 256, 256, 0, stream>>>(ybuf, xin, gamma, beta,
                                                             stats, out);
}

extern "C" void kernel_launch(void* const* d_in, const int* in_sizes, int n_in,
                              void* d_out, int out_size, void* d_ws, size_t ws_size,
                              hipStream_t stream) {
    (void)in_sizes; (void)n_in; (void)out_size; (void)ws_size;

    const float* x   = (const float*)d_in[0];
    const float* w1  = (const float*)d_in[1];
    const float* b1  = (const float*)d_in[2];
    const float* ow1 = (const float*)d_in[3];
    const float* ob1 = (const float*)d_in[4];
    const float* g1  = (const float*)d_in[5];
    const float* be1 = (const float*)d_in[6];
    const float* w2  = (const float*)d_in[7];
    const float* b2  = (const float*)d_in[8];
    const float* ow2 = (const float*)d_in[9];
    const float* ob2 = (const float*)d_in[10];
    const float* g2  = (const float*)d_in[11];
    const float* be2 = (const float*)d_in[12];

    float* out = (float*)d_out;

    // Workspace layout (floats): off | y | wTi | owTi | partials | stats
    float* ws       = (float*)d_ws;
    float* off      = ws;                               // B*18*HW = 1,179,648
    float* ybuf     = off + (size_t)BB * OFFC * HWSZ;   // B*C*H*W = 4,194,304
    float* wTi      = ybuf + (size_t)BB * CHW;          // 576*64 = 36,864
    float* owTi     = wTi + (size_t)KDIM * CC;          // 576*32 = 18,432
    float* partials = owTi + (size_t)KDIM * 32;         // 1,024
    float* stats    = partials + 1024;                  // 8

    // Block 1: x -> out
    run_block(x, w1, b1, ow1, ob1, g1, be1, off, ybuf, wTi, owTi,
              partials, stats, out, stream);
    // Block 2: out -> out (elementwise-safe in-place for the final apply)
    run_block(out, w2, b2, ow2, ob2, g2, be2, off, ybuf, wTi, owTi,
              partials, stats, out, stream);
}